// GCMCLayer_82403242541305
// MI455X (gfx1250) — compile-verified
//
#include <hip/hip_runtime.h>
#include <hip/hip_bf16.h>

// Problem constants (from reference)
#define R_    5
#define NU_   100000
#define NI_   50000
#define E_    1000000
#define IND_  64
#define MSG_  32
#define OUTC_ (R_ * MSG_)   // 160

typedef float v2f __attribute__((ext_vector_type(2)));
typedef float v8f __attribute__((ext_vector_type(8)));

// ---------------------------------------------------------------------------
// Kernel 1: per-rating fp32 GEMM  x[r] = (feat @ W[r]) * sqrt_deg  via
// V_WMMA_F32_16X16X4_F32.  One wave produces a 16x32 tile (2 N-tiles of 16).
// B fragments for the whole K=64 of W[r] are preloaded in registers and
// reused across all M-tiles the wave processes.
// ---------------------------------------------------------------------------
__global__ __launch_bounds__(256)
void gcmc_xform_wmma(const float* __restrict__ feat,   // [nrows, 64]
                     const float* __restrict__ W,      // [R, 64, 32]
                     const float* __restrict__ sdeg,   // [nrows]
                     float* __restrict__ x,            // [R, nrows, 32]
                     int nrows, int ntiles)
{
    const int r    = blockIdx.y;
    const int lane = threadIdx.x & 31;
    const int wave = threadIdx.x >> 5;
    const int mlo  = lane & 15;          // M (A) or N (B/C/D) index 0..15
    const int hi   = lane >> 4;          // 0: K=0,1 / M=v ; 1: K=2,3 / M=v+8
    const int khalf = hi * 2;

    const float* Wr = W + (size_t)r * IND_ * MSG_;

    // Preload B fragments: b[nt][ks] holds B[k0+khalf+{0,1}][nt*16 + mlo]
    v2f b0[16], b1[16];
#pragma unroll
    for (int ks = 0; ks < 16; ++ks) {
        const int k = ks * 4 + khalf;
        b0[ks].x = Wr[(k + 0) * MSG_ + mlo];
        b0[ks].y = Wr[(k + 1) * MSG_ + mlo];
        b1[ks].x = Wr[(k + 0) * MSG_ + 16 + mlo];
        b1[ks].y = Wr[(k + 1) * MSG_ + 16 + mlo];
    }

    for (int tile = blockIdx.x * 8 + wave; tile < ntiles; tile += gridDim.x * 8) {
        const int row0 = tile * 16;
        v8f c0 = {};
        v8f c1 = {};
#pragma unroll
        for (int ks = 0; ks < 16; ++ks) {
            const int k = ks * 4 + khalf;
            v2f a;
            const float* ap = feat + (size_t)(row0 + mlo) * IND_ + k;
            a.x = ap[0];
            a.y = ap[1];
            // (neg_a, A, neg_b, B, c_mod, C, reuse_a, reuse_b)
            c0 = __builtin_amdgcn_wmma_f32_16x16x4_f32(false, a, false, b0[ks],
                                                       (short)0, c0, false, false);
            c1 = __builtin_amdgcn_wmma_f32_16x16x4_f32(false, a, false, b1[ks],
                                                       (short)0, c1, false, false);
        }
        // Store tile, pre-scaled by sqrt_deg[row].
        float* xr = x + ((size_t)r * nrows + row0) * MSG_;
#pragma unroll
        for (int v = 0; v < 8; ++v) {
            const int m  = v + (hi << 3);
            const float sd = sdeg[row0 + m];
            xr[(size_t)m * MSG_ + mlo]      = c0[v] * sd;
            xr[(size_t)m * MSG_ + 16 + mlo] = c1[v] * sd;
        }
    }
}

// ---------------------------------------------------------------------------
// Kernel 2: edge scatter.  One wave32 per edge, lane = message feature.
// Both directions handled per edge:
//   out_i[dst, r*32+m] += x_u[r, src, m]
//   out_u[src, r*32+m] += x_i[r, dst, m]
// Hardware non-returning fp32 atomic (global_atomic_add_f32, device scope:
// RMW resolves at L2, the coherence point; no return value -> STOREcnt path).
// ---------------------------------------------------------------------------
__device__ __forceinline__ void atomic_add_f32(float* p, float v) {
    asm volatile("global_atomic_add_f32 %0, %1, off scope:SCOPE_DEV"
                 :
                 : "v"(p), "v"(v)
                 : "memory");
}

__global__ __launch_bounds__(256)
void gcmc_scatter(const int* __restrict__ esrc,  // [R, E]
                  const int* __restrict__ edst,  // [R, E]
                  const float* __restrict__ xu,  // [R, NU, 32] (pre-scaled)
                  const float* __restrict__ xi,  // [R, NI, 32] (pre-scaled)
                  float* __restrict__ out_u,     // [NU, 160]
                  float* __restrict__ out_i)     // [NI, 160]
{
    const int r    = blockIdx.y;
    const int lane = threadIdx.x & 31;
    // Wave-uniform edge id -> index loads become scalar (SMEM) loads.
    const int wave = __builtin_amdgcn_readfirstlane((int)(threadIdx.x >> 5));
    const size_t eidx = (size_t)r * E_ + (size_t)blockIdx.x * 8 + wave;

    const int s = esrc[eidx];
    const int d = edst[eidx];

    const float vu = xu[((size_t)r * NU_ + s) * MSG_ + lane];
    const float vi = xi[((size_t)r * NI_ + d) * MSG_ + lane];

    atomic_add_f32(&out_i[(size_t)d * OUTC_ + r * MSG_ + lane], vu);
    atomic_add_f32(&out_u[(size_t)s * OUTC_ + r * MSG_ + lane], vi);
}

// ---------------------------------------------------------------------------
// Kernel 3: finalize: out = relu(out * sqrt_deg[row]), in place, float4
// vectorized.  160 cols per row = 40 float4 per row.
// ---------------------------------------------------------------------------
__global__ __launch_bounds__(256)
void gcmc_finalize(float4* __restrict__ out,
                   const float* __restrict__ sdu,
                   const float* __restrict__ sdi)
{
    const size_t idx    = (size_t)blockIdx.x * 256 + threadIdx.x;  // float4 idx
    const size_t totalU = (size_t)NU_ * (OUTC_ / 4);
    const size_t total  = totalU + (size_t)NI_ * (OUTC_ / 4);
    if (idx >= total) return;
    const float sd = (idx < totalU) ? sdu[idx / (OUTC_ / 4)]
                                    : sdi[(idx - totalU) / (OUTC_ / 4)];
    float4 v = out[idx];
    v.x = fmaxf(v.x * sd, 0.0f);
    v.y = fmaxf(v.y * sd, 0.0f);
    v.z = fmaxf(v.z * sd, 0.0f);
    v.w = fmaxf(v.w * sd, 0.0f);
    out[idx] = v;
}

// ---------------------------------------------------------------------------
// Launch
// inputs: 0 ufeat [NU,64], 1 ifeat [NI,64], 2 W_u [R,64,32], 3 W_i [R,64,32],
//         4 sqrt_deg_u [NU,1], 5 sqrt_deg_i [NI,1], 6 edge_src [R,E],
//         7 edge_dst [R,E]
// output: u_out [NU,160] then i_out [NI,160], fp32
// ws:     x_u (R*NU*32 f32 = 64MB) then x_i (R*NI*32 f32 = 32MB)
// ---------------------------------------------------------------------------
extern "C" void kernel_launch(void* const* d_in, const int* in_sizes, int n_in,
                              void* d_out, int out_size, void* d_ws, size_t ws_size,
                              hipStream_t stream) {
    const float* ufeat = (const float*)d_in[0];
    const float* ifeat = (const float*)d_in[1];
    const float* W_u   = (const float*)d_in[2];
    const float* W_i   = (const float*)d_in[3];
    const float* sdu   = (const float*)d_in[4];
    const float* sdi   = (const float*)d_in[5];
    const int*   esrc  = (const int*)d_in[6];
    const int*   edst  = (const int*)d_in[7];

    float* out_u = (float*)d_out;
    float* out_i = out_u + (size_t)NU_ * OUTC_;

    float* xu = (float*)d_ws;
    float* xi = xu + (size_t)R_ * NU_ * MSG_;

    // Zero the accumulation target (graph-capture safe).
    hipMemsetAsync(d_out, 0, (size_t)out_size * sizeof(float), stream);

    // Per-rating transforms (WMMA fp32). 100000/16 = 6250, 50000/16 = 3125.
    gcmc_xform_wmma<<<dim3(80, R_), 256, 0, stream>>>(ufeat, W_u, sdu, xu,
                                                      NU_, NU_ / 16);
    gcmc_xform_wmma<<<dim3(40, R_), 256, 0, stream>>>(ifeat, W_i, sdi, xi,
                                                      NI_, NI_ / 16);

    // Edge scatter: 8 edges per 256-thread block; E divisible by 8.
    gcmc_scatter<<<dim3(E_ / 8, R_), 256, 0, stream>>>(esrc, edst, xu, xi,
                                                       out_u, out_i);

    // Scale + ReLU (float4).
    const size_t total4 = ((size_t)NU_ + NI_) * (OUTC_ / 4);
    gcmc_finalize<<<(unsigned)((total4 + 255) / 256), 256, 0, stream>>>(
        (float4*)d_out, sdu, sdi);
}